// Top2Router_75144747811318
// MI455X (gfx1250) — compile-verified
//
#include <hip/hip_runtime.h>
#include <hip/hip_bf16.h>
#include <math.h>

// Problem constants (match reference)
#define D_MODEL   2048
#define N_EXPERTS 64
#define TOKENS    16384          // 4 * 4096
#define TOK_PER_BLK 16

// d_out flat layout (reference tuple, concatenated):
//   expert_probs   [4,4096,2]  -> offset 0,        32768 floats
//   expert_indices [4,4096,2]  -> offset 32768,    32768 floats (indices as float)
//   expert_mask    [4,4096,64] -> offset 65536,  1048576 floats
//   aux_load_loss  scalar      -> offset 1114112
//   router_entropy scalar      -> offset 1114113
#define OFF_PROBS 0
#define OFF_IDX   32768
#define OFF_MASK  65536
#define OFF_AUX   1114112
#define OFF_ENT   1114113

typedef __attribute__((ext_vector_type(2))) float v2f;
typedef __attribute__((ext_vector_type(8))) float v8f;

// ---------------------------------------------------------------------------
// Kernel 0: zero the workspace accumulators (importance[64], load[64], ent[1])
// ---------------------------------------------------------------------------
__global__ void router_zero_ws(float* __restrict__ ws) {
  int i = threadIdx.x;
  if (i < 129) ws[i] = 0.0f;
}

// ---------------------------------------------------------------------------
// Kernel 1: fused gating GEMM (WMMA f32 16x16x4) + softmax + top2 + mask
//           + partial reductions into ws via atomics.
// Block = 128 threads = 4 waves. Wave w computes experts [16w, 16w+16) for
// the block's 16 tokens. Grid = TOKENS/16 = 1024 blocks.
// ---------------------------------------------------------------------------
__global__ __launch_bounds__(128) void router_main(
    const float* __restrict__ x,     // [TOKENS, D_MODEL]
    const float* __restrict__ W,     // [N_EXPERTS, D_MODEL]
    const float* __restrict__ temp,  // [1]
    float* __restrict__ out,
    float* __restrict__ ws)
{
  __shared__ float lds_row[TOK_PER_BLK * N_EXPERTS]; // logits, then normalized probs
  __shared__ int   lds_idx[TOK_PER_BLK * 2];
  __shared__ float lds_ent[TOK_PER_BLK];

  const int tid  = threadIdx.x;
  const int lane = tid & 31;
  const int wave = tid >> 5;       // 0..3 -> expert tile base 16*wave
  const int half = lane >> 4;      // 0 or 1 (K sub-pair selector)
  const int l16  = lane & 15;      // row-within-tile (A: token, B: expert)
  const int t0   = blockIdx.x * TOK_PER_BLK;

  // A-operand (16x4 f32): lane L<16 holds row M=L, {K, K+1}; lane L>=16 holds
  // row M=L-16, {K+2, K+3}.  B-operand (4x16) mirrors this with N=lane row of W.
  const float* xrow = x + (size_t)(t0 + l16) * D_MODEL + 2 * half;
  const float* wrow = W + (size_t)(wave * 16 + l16) * D_MODEL + 2 * half;

  v8f c0 = {}, c1 = {}, c2 = {}, c3 = {};
  for (int kk = 0; kk < D_MODEL; kk += 16) {
    v2f a0 = *(const v2f*)(xrow + kk);
    v2f b0 = *(const v2f*)(wrow + kk);
    v2f a1 = *(const v2f*)(xrow + kk + 4);
    v2f b1 = *(const v2f*)(wrow + kk + 4);
    v2f a2 = *(const v2f*)(xrow + kk + 8);
    v2f b2 = *(const v2f*)(wrow + kk + 8);
    v2f a3 = *(const v2f*)(xrow + kk + 12);
    v2f b3 = *(const v2f*)(wrow + kk + 12);
    // 8 args: (neg_a, A, neg_b, B, c_mod, C, reuse_a, reuse_b)
    c0 = __builtin_amdgcn_wmma_f32_16x16x4_f32(false, a0, false, b0, (short)0, c0, false, false);
    c1 = __builtin_amdgcn_wmma_f32_16x16x4_f32(false, a1, false, b1, (short)0, c1, false, false);
    c2 = __builtin_amdgcn_wmma_f32_16x16x4_f32(false, a2, false, b2, (short)0, c2, false, false);
    c3 = __builtin_amdgcn_wmma_f32_16x16x4_f32(false, a3, false, b3, (short)0, c3, false, false);
  }

  // C/D layout: lane<16 -> (M=r, N=lane), lane>=16 -> (M=r+8, N=lane-16)
  #pragma unroll
  for (int r = 0; r < 8; ++r) {
    int m = r + 8 * half;
    int n = wave * 16 + l16;
    lds_row[m * N_EXPERTS + n] = c0[r] + c1[r] + c2[r] + c3[r];
  }
  __syncthreads();

  // ---- per-token epilogue: softmax(logits / t), top-2, entropy ----
  if (tid < TOK_PER_BLK) {
    float t = temp[0];
    t = fminf(fmaxf(t, 0.1f), 5.0f);
    float inv_t = 1.0f / t;
    float* row = &lds_row[tid * N_EXPERTS];

    float m = -3.402823466e+38f;
    #pragma unroll 8
    for (int e = 0; e < N_EXPERTS; ++e) m = fmaxf(m, row[e] * inv_t);
    float s = 0.0f;
    #pragma unroll 8
    for (int e = 0; e < N_EXPERTS; ++e) {
      float p = __expf(row[e] * inv_t - m);
      row[e] = p;
      s += p;
    }
    float inv_s = 1.0f / s;

    float p1 = -1.0f; int i1 = 0;
    #pragma unroll 8
    for (int e = 0; e < N_EXPERTS; ++e) {
      float p = row[e] * inv_s;
      row[e] = p;                       // store normalized probs back
      if (p > p1) { p1 = p; i1 = e; }
    }
    float p2 = -1.0f; int i2 = 0;
    #pragma unroll 8
    for (int e = 0; e < N_EXPERTS; ++e) {
      if (e == i1) continue;
      float p = row[e];
      if (p > p2) { p2 = p; i2 = e; }
    }

    float ent = 0.0f;
    #pragma unroll 8
    for (int e = 0; e < N_EXPERTS; ++e) {
      float p = row[e];
      ent -= p * __logf(fmaxf(p, 1e-8f));
    }

    int tok = t0 + tid;
    out[OFF_PROBS + tok * 2 + 0] = p1;
    out[OFF_PROBS + tok * 2 + 1] = p2;
    out[OFF_IDX   + tok * 2 + 0] = (float)i1;
    out[OFF_IDX   + tok * 2 + 1] = (float)i2;
    lds_idx[tid * 2 + 0] = i1;
    lds_idx[tid * 2 + 1] = i2;
    lds_ent[tid] = ent;
  }
  __syncthreads();

  // ---- per-expert phase: mask writes (coalesced) + ws accumulation ----
  if (tid < N_EXPERTS) {
    float imp = 0.0f;
    float cnt = 0.0f;
    #pragma unroll
    for (int tok = 0; tok < TOK_PER_BLK; ++tok) {
      int i1 = lds_idx[tok * 2 + 0];
      int i2 = lds_idx[tok * 2 + 1];
      float mval = ((tid == i1) || (tid == i2)) ? 1.0f : 0.0f;
      out[OFF_MASK + (size_t)(t0 + tok) * N_EXPERTS + tid] = mval;
      cnt += mval;
      imp += lds_row[tok * N_EXPERTS + tid];
    }
    atomicAdd(&ws[tid], imp);            // sum of probs per expert
    atomicAdd(&ws[N_EXPERTS + tid], cnt);// top-2 selection count per expert
  }
  if (tid == 0) {
    float ent = 0.0f;
    #pragma unroll
    for (int tok = 0; tok < TOK_PER_BLK; ++tok) ent += lds_ent[tok];
    atomicAdd(&ws[128], ent);
  }
}

// ---------------------------------------------------------------------------
// Kernel 2: finalize the two scalar losses. 1 block, 64 threads.
// ---------------------------------------------------------------------------
__global__ __launch_bounds__(64) void router_finalize(
    const float* __restrict__ ws, float* __restrict__ out)
{
  __shared__ float red[N_EXPERTS];
  int tid = threadIdx.x;
  const float Tf = (float)TOKENS;
  float importance = ws[tid] / Tf;                 // mean prob per expert
  float load       = ws[N_EXPERTS + tid] / (Tf + 1e-6f);
  red[tid] = importance * load;
  __syncthreads();
  for (int s = 32; s > 0; s >>= 1) {
    if (tid < s) red[tid] += red[tid + s];
    __syncthreads();
  }
  if (tid == 0) {
    out[OFF_AUX] = red[0] * (float)N_EXPERTS * 0.01f;  // aux_load_loss
    out[OFF_ENT] = (ws[128] / Tf) * 0.01f;             // router_entropy
  }
}

extern "C" void kernel_launch(void* const* d_in, const int* in_sizes, int n_in,
                              void* d_out, int out_size, void* d_ws, size_t ws_size,
                              hipStream_t stream) {
  const float* x    = (const float*)d_in[0];   // [4,4096,2048]
  const float* W    = (const float*)d_in[1];   // [64,2048]
  const float* temp = (const float*)d_in[2];   // [1]
  float* out = (float*)d_out;
  float* ws  = (float*)d_ws;

  router_zero_ws<<<1, 256, 0, stream>>>(ws);
  router_main<<<TOKENS / TOK_PER_BLK, 128, 0, stream>>>(x, W, temp, out, ws);
  router_finalize<<<1, 64, 0, stream>>>(ws, out);
}